// AMPSShare_26001732010119
// MI455X (gfx1250) — compile-verified
//
#include <hip/hip_runtime.h>

typedef __attribute__((ext_vector_type(2))) float v2f;
typedef __attribute__((ext_vector_type(4))) float v4f;
typedef __attribute__((ext_vector_type(8))) float v8f;

#define N_SITES 784
#define DBOND   10
#define BATCH   16384

// ---------------------------------------------------------------------------
// Prep kernel: pack A' = (tensors + I)^T into WMMA A-operand layout.
// Layout (v2f units): [site][ks(3)][lane(32)][ch(2)] -> {vgpr0, vgpr1}
//   lane<16 :  vgpr0 = A[l=4ks  , r=lane   , ch],  vgpr1 = A[4ks+1, lane   , ch]
//   lane>=16:  vgpr0 = A[l=4ks+2, r=lane-16, ch],  vgpr1 = A[4ks+3, lane-16, ch]
// Zero outside l<10, r<10 (K padded 10->12, M padded 10->16).
// Channel-contiguous so the main kernel loads one b128 per (site, ks).
// The zero K-padding (l=10,11) is what lets the main kernel skip zeroing the
// hi-lane halves of the bv2 B-operand.
// ---------------------------------------------------------------------------
__global__ void mps_prep_apack(const float* __restrict__ tensors,
                               float* __restrict__ apack) {
    int t = blockIdx.x * blockDim.x + threadIdx.x;   // ((site*3+ks)*32+lane)*2+ch
    if (t >= N_SITES * 3 * 32 * 2) return;
    int ch   = t & 1;
    int lane = (t >> 1) & 31;
    int ks   = (t >> 6) % 3;
    int site = t / 192;
    int r    = lane & 15;
    int k0   = ks * 4 + ((lane >> 4) & 1) * 2;

    float v0 = 0.0f, v1 = 0.0f;
    if (r < DBOND) {
        int l0 = k0, l1 = k0 + 1;
        if (l0 < DBOND)
            v0 = tensors[((site * DBOND + l0) * DBOND + r) * 2 + ch] + (l0 == r ? 1.0f : 0.0f);
        if (l1 < DBOND)
            v1 = tensors[((site * DBOND + l1) * DBOND + r) * 2 + ch] + (l1 == r ? 1.0f : 0.0f);
    }
    float* dst = apack + (size_t)t * 2;
    dst[0] = v0;
    dst[1] = v1;
}

// ---------------------------------------------------------------------------
// Main kernel: one wave32 per 16 batch rows, sequential chain over 784 sites.
// Per site: D0/D1 (16x16 f32) = Apack_ch^T x LV^T via 3x v_wmma_f32_16x16x4_f32
// each; logits sit in c[0] of lanes 0..15; two-way log-softmax reduced to
// -ln2*log2(1+exp2(dl*log2e)); next-LV B-operand rebuilt with 4 SWAPX16
// ds_swizzles (each carries payload in both directions).
// ---------------------------------------------------------------------------
__global__ __launch_bounds__(256) void mps_chain_wmma(
    const float* __restrict__ data,      // (BATCH, N_SITES) in {0,1}
    const float* __restrict__ apack,     // packed WMMA A operands
    float* __restrict__ out) {           // (BATCH,)
    const int lane  = threadIdx.x & 31;
    const int wave  = (blockIdx.x * blockDim.x + threadIdx.x) >> 5;
    const int bbase = wave * 16;
    const bool lo   = lane < 16;

    // lanes b and b+16 both read batch row (lane & 15); rows are 16B aligned
    const float* drow = data + (size_t)(bbase + (lane & 15)) * N_SITES;

    // LV state in B-operand layout (N=batch=lane%16, K rows split lo/hi half):
    //  bv0 = {LV[b,0]|LV[b,2], LV[b,1]|LV[b,3]}
    //  bv1 = {LV[b,4]|LV[b,6], LV[b,5]|LV[b,7]}
    //  bv2 = {LV[b,8]|dontcare, LV[b,9]|dontcare}  (k=10,11 hit zero A columns)
    // init lv = e0 (reproduces the reference's site-0 special case exactly)
    v2f bv0, bv1, bv2;
    bv0.x = lo ? 1.0f : 0.0f; bv0.y = 0.0f;
    bv1.x = 0.0f; bv1.y = 0.0f;
    bv2.x = 0.0f; bv2.y = 0.0f;

    float acc = 0.0f;

    for (int s = 0; s < N_SITES; s += 4) {
        v4f d4 = *(const v4f*)(drow + s);
#pragma unroll
        for (int j = 0; j < 4; ++j) {
            const v4f* a = (const v4f*)apack + (size_t)(s + j) * 96 + lane;
            v4f A0 = a[0];    // ks0: {ch0.x, ch0.y, ch1.x, ch1.y}
            v4f A1 = a[32];   // ks1
            v4f A2 = a[64];   // ks2
            float d = d4[j];

            v2f a00; a00.x = A0.x; a00.y = A0.y;
            v2f a01; a01.x = A0.z; a01.y = A0.w;
            v2f a10; a10.x = A1.x; a10.y = A1.y;
            v2f a11; a11.x = A1.z; a11.y = A1.w;
            v2f a20; a20.x = A2.x; a20.y = A2.y;
            v2f a21; a21.x = A2.z; a21.y = A2.w;

            v8f c0 = {}, c1 = {};
            c0 = __builtin_amdgcn_wmma_f32_16x16x4_f32(false, a00, false, bv0, (short)0, c0, false, false);
            c0 = __builtin_amdgcn_wmma_f32_16x16x4_f32(false, a10, false, bv1, (short)0, c0, false, false);
            c0 = __builtin_amdgcn_wmma_f32_16x16x4_f32(false, a20, false, bv2, (short)0, c0, false, false);
            c1 = __builtin_amdgcn_wmma_f32_16x16x4_f32(false, a01, false, bv0, (short)0, c1, false, false);
            c1 = __builtin_amdgcn_wmma_f32_16x16x4_f32(false, a11, false, bv1, (short)0, c1, false, false);
            c1 = __builtin_amdgcn_wmma_f32_16x16x4_f32(false, a21, false, bv2, (short)0, c1, false, false);

            bool pick0 = (d != 0.0f);

            // channel select: n[i] = u_sel[b, r=i (+8 on hi lanes)]
            float n0 = pick0 ? c0[0] : c1[0];
            float n1 = pick0 ? c0[1] : c1[1];
            float n2 = pick0 ? c0[2] : c1[2];
            float n3 = pick0 ? c0[3] : c1[3];
            float n4 = pick0 ? c0[4] : c1[4];
            float n5 = pick0 ? c0[5] : c1[5];
            float n6 = pick0 ? c0[6] : c1[6];
            float n7 = pick0 ? c0[7] : c1[7];

            // two-way log-softmax (valid in lanes 0..15 where c[0]=u[b,0]):
            // chosen = n0; logp = -ln2 * log2(1 + exp2((other-chosen)*log2e))
            float oth = pick0 ? c1[0] : c0[0];
            float tt  = __builtin_amdgcn_exp2f((oth - n0) * 1.44269504088896f);
            float lg  = __builtin_amdgcn_logf(1.0f + tt);
            acc = __builtin_fmaf(-0.693147180559945f, lg, acc);

            // SWAPX16 lane exchange (and=0x1f, or=0, xor=0x10 -> imm 0x401F).
            // Each swizzle carries lo->hi and hi->lo payload simultaneously:
            //   lo needs partner r8,r9 (= hi n0,n1); hi needs partner n2,n3,n6,n7
            float t1 = lo ? n2 : n0;
            float t2 = lo ? n3 : n1;
            float s1 = __int_as_float(__builtin_amdgcn_ds_swizzle(__float_as_int(t1), 0x401F));
            float s2 = __int_as_float(__builtin_amdgcn_ds_swizzle(__float_as_int(t2), 0x401F));
            float s3 = __int_as_float(__builtin_amdgcn_ds_swizzle(__float_as_int(n6), 0x401F));
            float s4 = __int_as_float(__builtin_amdgcn_ds_swizzle(__float_as_int(n7), 0x401F));

            bv0.x = lo ? n0 : s1;   // LV[b,0] | LV[b,2]
            bv0.y = lo ? n1 : s2;   // LV[b,1] | LV[b,3]
            bv1.x = lo ? n4 : s3;   // LV[b,4] | LV[b,6]
            bv1.y = lo ? n5 : s4;   // LV[b,5] | LV[b,7]
            bv2.x = s1;             // LV[b,8] | (x: multiplied by zero A col)
            bv2.y = s2;             // LV[b,9] | (x: multiplied by zero A col)
        }
    }

    if (lo) out[bbase + lane] = acc;
}

// ---------------------------------------------------------------------------
extern "C" void kernel_launch(void* const* d_in, const int* in_sizes, int n_in,
                              void* d_out, int out_size, void* d_ws, size_t ws_size,
                              hipStream_t stream) {
    const float* data    = (const float*)d_in[0];   // (BATCH, N_SITES) f32
    const float* tensors = (const float*)d_in[1];   // (N_SITES, D, D, 2) f32
    float* out   = (float*)d_out;                   // (BATCH,) f32
    float* apack = (float*)d_ws;                    // 784*192*2 floats = 1.2 MB

    int prep_threads = N_SITES * 3 * 32 * 2;
    mps_prep_apack<<<(prep_threads + 255) / 256, 256, 0, stream>>>(tensors, apack);

    int waves = BATCH / 16;                         // 1024 waves
    int threads = waves * 32;                       // 32768 threads
    mps_chain_wmma<<<threads / 256, 256, 0, stream>>>(data, apack, out);
}